// GCN_19885698580718
// MI455X (gfx1250) — compile-verified
//
#include <hip/hip_runtime.h>
#include <cstdint>

// GCN link prediction for MI455X (gfx1250, wave32).
//
// Shapes: N=100000 nodes, IN=128, HID=OUT=64, E=1.6M edges, L=500K label edges.
// Memory-bound: ~2 GB of gather/scatter traffic, but the 77 MB working set fits
// in the 192 MB L2. GEMMs (2.5 GFLOP) run on the f32 WMMA pipe at full precision.

typedef float v2f __attribute__((ext_vector_type(2)));
typedef float v8f __attribute__((ext_vector_type(8)));

#define THREADS 256

// ---------------- degree / normalization ----------------------------------

__global__ void k_init_deg(int* __restrict__ deg, int N) {
  int i = blockIdx.x * THREADS + threadIdx.x;
  if (i < N) deg[i] = 1;  // self-loop contributes 1 to in-degree
}

__global__ void k_count_deg(const int* __restrict__ dst, int E, int* __restrict__ deg) {
  int e = blockIdx.x * THREADS + threadIdx.x;
  if (e < E) atomicAdd(&deg[dst[e]], 1);
}

__global__ void k_calc_dis(const int* __restrict__ deg, float* __restrict__ dis, int N) {
  int i = blockIdx.x * THREADS + threadIdx.x;
  if (i < N) dis[i] = rsqrtf((float)deg[i]);  // deg >= 1 always (self-loop)
}

// ---------------- WMMA f32 GEMM:  H[M,64] = (relu?)A[M,K] @ W[K,64] --------
//
// One wave per 16-row M tile, covering all 64 output columns with four 16x16
// accumulators. V_WMMA_F32_16X16X4_F32 VGPR layouts (ISA 7.12.2):
//   A 16x4:  lane = 16*(k>=2) + m,  vgpr v -> K = 2*(lane>>4) + v
//   B 4x16:  lane = 16*(k>=2) + n,  vgpr v -> K = 2*(lane>>4) + v
//   C/D:     vgpr r, lanes 0-15 -> M=r, N=lane; lanes 16-31 -> M=r+8, N=lane-16
//
// W is staged in LDS K-pair interleaved:  sW[(k>>1)*128 + n*2 + (k&1)]
// so each B operand is ONE aligned ds_load_b64 (both K values adjacent),
// eliminating the v_dual_mov glue the split-word layout required.
template <int K, bool RELU>
__global__ __launch_bounds__(THREADS) void k_gemm_wmma(
    const float* __restrict__ A, const float* __restrict__ W,
    float* __restrict__ H, int M) {
  __shared__ float sW[K * 64];
  for (int i = threadIdx.x; i < K * 64; i += THREADS) {
    int k = i >> 6, n = i & 63;
    sW[(k >> 1) * 128 + n * 2 + (k & 1)] = W[i];
  }
  __syncthreads();

  const int lane  = threadIdx.x & 31;
  const int wave  = threadIdx.x >> 5;
  const int tile  = blockIdx.x * (THREADS / 32) + wave;
  if (tile * 16 >= M) return;              // wave-uniform: EXEC stays all-ones

  const int mbase = tile * 16;
  const int nl    = lane & 15;
  const int khalf = lane >> 4;
  const float* arow = A + (size_t)(mbase + nl) * K;  // lane's A row (m = lane&15)

  v8f c0 = {}, c1 = {}, c2 = {}, c3 = {};

  for (int k0 = 0; k0 < K; k0 += 4) {
    const int ka = k0 + 2 * khalf;         // this lane-half's K pair
    v2f a = *(const v2f*)(arow + ka);
    if (RELU) { a.x = fmaxf(a.x, 0.0f); a.y = fmaxf(a.y, 0.0f); }

    // one b64 LDS load per n-tile: {W[ka][n], W[ka+1][n]}
    const v2f* wp = (const v2f*)(sW + (ka >> 1) * 128 + nl * 2);
    v2f b0 = wp[0];
    v2f b1 = wp[16];
    v2f b2 = wp[32];
    v2f b3 = wp[48];
    c0 = __builtin_amdgcn_wmma_f32_16x16x4_f32(false, a, false, b0, (short)0, c0, false, false);
    c1 = __builtin_amdgcn_wmma_f32_16x16x4_f32(false, a, false, b1, (short)0, c1, false, false);
    c2 = __builtin_amdgcn_wmma_f32_16x16x4_f32(false, a, false, b2, (short)0, c2, false, false);
    c3 = __builtin_amdgcn_wmma_f32_16x16x4_f32(false, a, false, b3, (short)0, c3, false, false);
  }

  // Store D: lane writes M = mbase + r + 8*khalf, N = nl + 16*t
  float* hp = H + (size_t)(mbase + 8 * khalf) * 64 + nl;
#pragma unroll
  for (int r = 0; r < 8; ++r) {
    hp[(size_t)r * 64 + 0]  = c0[r];
    hp[(size_t)r * 64 + 16] = c1[r];
    hp[(size_t)r * 64 + 32] = c2[r];
    hp[(size_t)r * 64 + 48] = c3[r];
  }
}

// ---------------- aggregation ---------------------------------------------

// z[i,:] = bias + h[i,:] * dis[i]^2   (self-loop term; also initializes buffer)
__global__ __launch_bounds__(THREADS) void k_init_agg(
    const float* __restrict__ h, const float* __restrict__ dis,
    const float* __restrict__ bias, float* __restrict__ z, int N) {
  int i = blockIdx.x * THREADS + threadIdx.x;  // over N*32 (2 channels/thread)
  int node = i >> 5;
  int c = (i & 31) * 2;
  if (node < N) {
    float d  = dis[node];
    float d2 = d * d;
    size_t o = (size_t)node * 64 + c;
    z[o]     = bias[c]     + h[o]     * d2;
    z[o + 1] = bias[c + 1] + h[o + 1] * d2;
  }
}

// z[dst,:] += h[src,:] * dis[src]*dis[dst]  — one wave per edge, float2/lane
__global__ __launch_bounds__(THREADS) void k_edge_agg(
    const int* __restrict__ src, const int* __restrict__ dst,
    const float* __restrict__ dis, const float* __restrict__ h,
    float* __restrict__ z, int E) {
  int e = blockIdx.x * (THREADS / 32) + (threadIdx.x >> 5);
  if (e >= E) return;
  int lane = threadIdx.x & 31;
  int s = src[e], d = dst[e];
  float norm = dis[s] * dis[d];
  v2f hv = *(const v2f*)(h + (size_t)s * 64 + lane * 2);
  float* zp = z + (size_t)d * 64 + lane * 2;
  unsafeAtomicAdd(zp,     hv.x * norm);   // hardware global_atomic_add_f32
  unsafeAtomicAdd(zp + 1, hv.y * norm);
}

// ---------------- decode: out[e] = dot(z[a[e]], z[b[e]]) -------------------

__global__ __launch_bounds__(THREADS) void k_decode(
    const int* __restrict__ ia, const int* __restrict__ ib,
    const float* __restrict__ z, float* __restrict__ out, int L) {
  int e = blockIdx.x * (THREADS / 32) + (threadIdx.x >> 5);
  if (e >= L) return;
  int lane = threadIdx.x & 31;
  int i = ia[e], j = ib[e];
  v2f p = *(const v2f*)(z + (size_t)i * 64 + lane * 2);
  v2f q = *(const v2f*)(z + (size_t)j * 64 + lane * 2);
  float s = p.x * q.x + p.y * q.y;
#pragma unroll
  for (int off = 16; off > 0; off >>= 1) s += __shfl_xor(s, off, 32);
  if (lane == 0) out[e] = s;
}

// ---------------- host orchestration --------------------------------------

extern "C" void kernel_launch(void* const* d_in, const int* in_sizes, int n_in,
                              void* d_out, int out_size, void* d_ws, size_t ws_size,
                              hipStream_t stream) {
  const float* x   = (const float*)d_in[0];
  const int*   ei  = (const int*)d_in[1];
  const int*   eli = (const int*)d_in[2];
  const float* W1  = (const float*)d_in[3];
  const float* b1  = (const float*)d_in[4];
  const float* W2  = (const float*)d_in[5];
  const float* b2  = (const float*)d_in[6];
  float* out = (float*)d_out;

  const int N = in_sizes[0] / 128;
  const int E = in_sizes[1] / 2;
  const int L = in_sizes[2] / 2;
  const int* srcv = ei;          // edge_index[0]
  const int* dstv = ei + E;      // edge_index[1]
  const int* la   = eli;
  const int* lb   = eli + L;

  // workspace carve-out (256B aligned): deg, dis, h, z1, z2  (~77 MB)
  char* ws = (char*)d_ws;
  size_t off = 0;
  auto carve = [&](size_t bytes) -> void* {
    void* p = ws + off;
    off += (bytes + 255) & ~(size_t)255;
    return p;
  };
  int*   deg = (int*)carve((size_t)N * 4);
  float* dis = (float*)carve((size_t)N * 4);
  float* h   = (float*)carve((size_t)N * 64 * 4);  // h1, later reused as h2
  float* z1  = (float*)carve((size_t)N * 64 * 4);
  float* z2  = (float*)carve((size_t)N * 64 * 4);

  const int bN  = (N + THREADS - 1) / THREADS;
  const int bE  = (E + THREADS - 1) / THREADS;
  const int bNA = ((N * 32) + THREADS - 1) / THREADS;
  const int mtiles = (N + 15) / 16;
  const int bG  = (mtiles + 7) / 8;          // 8 waves (M-tiles) per block
  const int bEw = (E + 7) / 8;               // 8 edges (waves) per block
  const int bLw = (L + 7) / 8;

  // normalization
  k_init_deg<<<bN, THREADS, 0, stream>>>(deg, N);
  k_count_deg<<<bE, THREADS, 0, stream>>>(dstv, E, deg);
  k_calc_dis<<<bN, THREADS, 0, stream>>>(deg, dis, N);

  // layer 1
  k_gemm_wmma<128, false><<<bG, THREADS, 0, stream>>>(x, W1, h, N);
  k_init_agg<<<bNA, THREADS, 0, stream>>>(h, dis, b1, z1, N);
  k_edge_agg<<<bEw, THREADS, 0, stream>>>(srcv, dstv, dis, h, z1, E);

  // layer 2 (ReLU fused into GEMM A-load; h buffer reused for h2)
  k_gemm_wmma<64, true><<<bG, THREADS, 0, stream>>>(z1, W2, h, N);
  k_init_agg<<<bNA, THREADS, 0, stream>>>(h, dis, b2, z2, N);
  k_edge_agg<<<bEw, THREADS, 0, stream>>>(srcv, dstv, dis, h, z2, E);

  // decode
  k_decode<<<bLw, THREADS, 0, stream>>>(la, lb, z2, out, L);
}